// GIN_Triplet_unit_21114059227223
// MI455X (gfx1250) — compile-verified
//
#include <hip/hip_runtime.h>

#define D 128

typedef __attribute__((ext_vector_type(2))) float v2f;
typedef __attribute__((ext_vector_type(8))) float v8f;

// ---------------------------------------------------------------------------
// Kernel 1: zero the scatter accumulator [N*D] and degree counters [N]
// ---------------------------------------------------------------------------
__global__ void zero_ws_kernel(float* __restrict__ summed,
                               float* __restrict__ cnt,
                               int n_nodes) {
  int i = blockIdx.x * blockDim.x + threadIdx.x;
  int total = n_nodes * D;
  if (i < total) summed[i] = 0.0f;
  if (i < n_nodes) cnt[i] = 0.0f;
}

// ---------------------------------------------------------------------------
// Kernel 2: per-edge message + scatter-add (one wave32 per edge, 4 f32/lane)
// msg[d] = relu(hidden[src][d] + ea[e]*We[d] + be[d]);  summed[dst] += msg
// ---------------------------------------------------------------------------
__global__ void scatter_msgs_kernel(const float* __restrict__ hidden,
                                    const long long* __restrict__ ei,   // [2,E] int64
                                    const float* __restrict__ ea,       // [E]
                                    const float* __restrict__ We,       // [D]
                                    const float* __restrict__ be,       // [D]
                                    float* __restrict__ summed,
                                    float* __restrict__ cnt,
                                    int n_edges) {
  int gwave = (blockIdx.x * blockDim.x + threadIdx.x) >> 5;
  int lane  = threadIdx.x & 31;
  if (gwave >= n_edges) return;

  int src = (int)ei[gwave];
  int dst = (int)ei[(size_t)n_edges + gwave];
  float a = ea[gwave];

  int d0 = lane * 4;
  float4 h = *(const float4*)(hidden + (size_t)src * D + d0);
  float4 w = *(const float4*)(We + d0);
  float4 b = *(const float4*)(be + d0);

  float m0 = fmaxf(h.x + fmaf(a, w.x, b.x), 0.0f);
  float m1 = fmaxf(h.y + fmaf(a, w.y, b.y), 0.0f);
  float m2 = fmaxf(h.z + fmaf(a, w.z, b.z), 0.0f);
  float m3 = fmaxf(h.w + fmaf(a, w.w, b.w), 0.0f);

  float* sp = summed + (size_t)dst * D + d0;
  atomicAdd(sp + 0, m0);
  atomicAdd(sp + 1, m1);
  atomicAdd(sp + 2, m2);
  atomicAdd(sp + 3, m3);
  if (lane == 0) atomicAdd(cnt + dst, 1.0f);
}

// ---------------------------------------------------------------------------
// Kernels 3+4: 16x16-tile fp32 WMMA GEMM, Y = act(X @ W + bias)
//   FUSE=1: X computed on the fly as (1+eps)*hidden + summed/max(cnt,1)
//   Each wave: 16 rows x all 128 cols (8 accumulator tiles), K in steps of 4.
//   W staged into 64KB LDS with gfx1250 async global->LDS b128 copies.
// ---------------------------------------------------------------------------
template <int FUSE, int RELU>
__launch_bounds__(256)
__global__ void gin_mlp_gemm_kernel(const float* __restrict__ Xa,   // hidden (fuse) or h
                                    const float* __restrict__ Xb,   // summed (fuse only)
                                    const float* __restrict__ cnt,  // fuse only
                                    const float* __restrict__ epsp, // fuse only
                                    const float* __restrict__ W,    // [D,D] row-major [in][out]
                                    const float* __restrict__ bias, // [D]
                                    float* __restrict__ Y,
                                    int n_rows) {
  __shared__ float Ws[D * D]; // 64 KB, row-major: Ws[k*D + col]

  // --- Stage W into LDS via async global->LDS (ASYNCcnt-tracked, no VGPR hop)
  {
    // LDS aperture: generic-pointer low 32 bits == LDS byte offset (ISA 10.2)
    unsigned lds_base = (unsigned)(uintptr_t)(&Ws[0]);
    const char* gbase = (const char*)W;
    // D*D floats = 4096 x 16B chunks; 256 threads x 16 iters
    for (int i = threadIdx.x; i < (D * D) / 4; i += blockDim.x) {
      unsigned lds_addr = lds_base + (unsigned)i * 16u;
      const void* gaddr = gbase + (size_t)i * 16u;
      asm volatile("global_load_async_to_lds_b128 %0, %1, off"
                   :: "v"(lds_addr), "v"(gaddr)
                   : "memory");
    }
    asm volatile("s_wait_asynccnt 0" ::: "memory");
  }
  __syncthreads();

  int wave = threadIdx.x >> 5;
  int lane = threadIdx.x & 31;
  int row0 = (blockIdx.x * 8 + wave) * 16;
  if (row0 >= n_rows) return;              // uniform per wave: EXEC stays all-1

  int m  = lane & 15;                      // A: row within tile; B/C/D: column
  int kh = lane >> 4;                      // K half for A/B fragments
  int row = row0 + m;
  if (row >= n_rows) row = n_rows - 1;     // safe clamp (N%16==0 so unused)

  float scale = 1.0f, inv = 1.0f;
  if (FUSE) {
    scale = 1.0f + epsp[0];
    inv = 1.0f / fmaxf(cnt[row], 1.0f);
  }

  v8f acc[8] = {};

  const float* xrow = Xa + (size_t)row * D;
  const float* srow = FUSE ? (Xb + (size_t)row * D) : nullptr;

#pragma unroll 2
  for (int ks = 0; ks < 32; ++ks) {
    int kbase = ks * 4 + kh * 2;           // A/B fragment: K = kbase, kbase+1
    v2f a;
    if (FUSE) {
      a.x = scale * xrow[kbase + 0] + srow[kbase + 0] * inv;
      a.y = scale * xrow[kbase + 1] + srow[kbase + 1] * inv;
    } else {
      a = *(const v2f*)(xrow + kbase);
    }
#pragma unroll
    for (int t = 0; t < 8; ++t) {
      int col = t * 16 + m;
      v2f b;
      b.x = Ws[(kbase + 0) * D + col];
      b.y = Ws[(kbase + 1) * D + col];
      acc[t] = __builtin_amdgcn_wmma_f32_16x16x4_f32(
          /*neg_a=*/false, a, /*neg_b=*/false, b,
          /*c_mod=*/(short)0, acc[t], /*reuse_a=*/false, /*reuse_b=*/false);
    }
  }

  // Epilogue: C/D layout -> element (vgpr v, lane): row = row0 + v + 8*kh,
  // col = t*16 + (lane&15)
#pragma unroll
  for (int t = 0; t < 8; ++t) {
    int col = t * 16 + m;
    float bv = bias[col];
#pragma unroll
    for (int v = 0; v < 8; ++v) {
      int r = row0 + v + 8 * kh;
      if (r < n_rows) {
        float val = acc[t][v] + bv;
        if (RELU) val = fmaxf(val, 0.0f);
        Y[(size_t)r * D + col] = val;
      }
    }
  }
}

// ---------------------------------------------------------------------------
extern "C" void kernel_launch(void* const* d_in, const int* in_sizes, int n_in,
                              void* d_out, int out_size, void* d_ws, size_t ws_size,
                              hipStream_t stream) {
  const float*      hidden = (const float*)d_in[0];
  const long long*  ei     = (const long long*)d_in[1];   // int64 [2,E]
  const float*      ea     = (const float*)d_in[2];       // [E,1]
  const float*      We     = (const float*)d_in[3];       // [1,D]
  const float*      be     = (const float*)d_in[4];       // [D]
  const float*      W1     = (const float*)d_in[5];       // [D,D]
  const float*      b1     = (const float*)d_in[6];
  const float*      W2     = (const float*)d_in[7];
  const float*      b2     = (const float*)d_in[8];
  const float*      eps    = (const float*)d_in[9];

  int n_nodes = in_sizes[0] / D;   // 50000
  int n_edges = in_sizes[2];       // 800000 (edge_attr flat count)
  float* out = (float*)d_out;

  // Workspace layout: summed [N*D] | h [N*D] | cnt [N]
  float* summed = (float*)d_ws;
  float* h      = summed + (size_t)n_nodes * D;
  float* cnt    = h + (size_t)n_nodes * D;

  int total = n_nodes * D;
  zero_ws_kernel<<<(total + 255) / 256, 256, 0, stream>>>(summed, cnt, n_nodes);

  // 8 edges per 256-thread block (1 wave32 per edge)
  scatter_msgs_kernel<<<(n_edges + 7) / 8, 256, 0, stream>>>(
      hidden, ei, ea, We, be, summed, cnt, n_edges);

  int row_tiles = (n_nodes + 15) / 16;      // 3125
  int blocks = (row_tiles + 7) / 8;         // 8 waves (row tiles) per block

  // Layer 1 (fused h0 computation + relu)
  gin_mlp_gemm_kernel<1, 1><<<blocks, 256, 0, stream>>>(
      hidden, summed, cnt, eps, W1, b1, h, n_nodes);

  // Layer 2
  gin_mlp_gemm_kernel<0, 0><<<blocks, 256, 0, stream>>>(
      h, nullptr, nullptr, nullptr, W2, b2, out, n_nodes);
}